// DGCNN_54013508714654
// MI455X (gfx1250) — compile-verified
//
#include <hip/hip_runtime.h>
#include <hip/hip_bf16.h>

typedef float v2f __attribute__((ext_vector_type(2)));
typedef float v8f __attribute__((ext_vector_type(8)));

#define NPTS   1024
#define BATCH  4
#define KNN    20
#define NROWS  (BATCH * NPTS)          // 4096
#define BNK    (BATCH * NPTS * KNN)    // 81920
#define FLT_BIG 3.402823466e38f

// ---------------------------------------------------------------------------
// Transpose input x (B, 8, N) -> X0 (B*N, 8) row-major
// ---------------------------------------------------------------------------
__global__ void k_transpose_in(const float* __restrict__ x, float* __restrict__ X0) {
    int i = blockIdx.x * 256 + threadIdx.x;          // over 4*8*1024
    if (i >= BATCH * 8 * NPTS) return;
    int n = i & (NPTS - 1);
    int c = (i >> 10) & 7;
    int b = i >> 13;
    X0[(b * NPTS + n) * 8 + c] = x[i];
}

// ---------------------------------------------------------------------------
// Row squared norms: xx[r] = sum_c X[r][c]^2
// ---------------------------------------------------------------------------
__global__ void k_row_norms(const float* __restrict__ X, float* __restrict__ xx, int C) {
    int r = blockIdx.x * 256 + threadIdx.x;
    if (r >= NROWS) return;
    const float* p = X + (size_t)r * C;
    float s = 0.f;
    for (int c = 0; c < C; ++c) s += p[c] * p[c];
    xx[r] = s;
}

// ---------------------------------------------------------------------------
// Gram + pairwise distance via V_WMMA_F32_16X16X4_F32.
// PD[b][n][m] = 2 * <x_n, x_m> - ||x_n||^2 - ||x_m||^2   (= -||x_n - x_m||^2)
// One wave computes a 16x16 tile. blockDim = (32, 4): 4 m-tiles per block.
// ---------------------------------------------------------------------------
__global__ void k_gram_pd(const float* __restrict__ X, const float* __restrict__ xx,
                          float* __restrict__ PD, int C) {
    const int lane = threadIdx.x;        // 0..31
    const int half = lane >> 4;          // 0/1
    const int lm   = lane & 15;
    const int b  = blockIdx.z;
    const int n0 = blockIdx.x * 16;
    const int m0 = (blockIdx.y * 4 + threadIdx.y) * 16;

    const float* Xb  = X + (size_t)b * NPTS * C;
    const float* xxb = xx + b * NPTS;
    const float* arow = Xb + (size_t)(n0 + lm) * C;   // A: row n, K split by lane half
    const float* brow = Xb + (size_t)(m0 + lm) * C;   // B: col m = lane%16

    v8f acc = {};
    for (int kk = 0; kk < C; kk += 4) {
        int k0 = kk + 2 * half;
        v2f a; a.x = arow[k0]; a.y = arow[k0 + 1];
        v2f bb; bb.x = brow[k0]; bb.y = brow[k0 + 1];
        acc = __builtin_amdgcn_wmma_f32_16x16x4_f32(false, a, false, bb,
                                                    (short)0, acc, false, false);
    }
    const float xm = xxb[m0 + lm];
#pragma unroll
    for (int r = 0; r < 8; ++r) {
        int n = n0 + r + 8 * half;
        PD[((size_t)(b * NPTS + n)) * NPTS + m0 + lm] = 2.0f * acc[r] - xxb[n] - xm;
    }
}

// ---------------------------------------------------------------------------
// Top-K (K=20) per row, descending value, lower index wins ties (jax top_k).
// One 256-thread block per row; iterative block-wide argmax.
// ---------------------------------------------------------------------------
__global__ void k_topk(const float* __restrict__ PD, int* __restrict__ IDX) {
    __shared__ float vals[NPTS];
    __shared__ float redv[256];
    __shared__ int   redi[256];
    const float* row = PD + (size_t)blockIdx.x * NPTS;
    const int t = threadIdx.x;
    for (int i = t; i < NPTS; i += 256) vals[i] = row[i];
    __syncthreads();
    for (int k = 0; k < KNN; ++k) {
        float bv = -FLT_BIG; int bi = NPTS;
        for (int i = t; i < NPTS; i += 256) {
            float v = vals[i];
            if (v > bv || (v == bv && i < bi)) { bv = v; bi = i; }
        }
        redv[t] = bv; redi[t] = bi;
        __syncthreads();
        for (int s = 128; s > 0; s >>= 1) {
            if (t < s) {
                float v2 = redv[t + s]; int i2 = redi[t + s];
                if (v2 > redv[t] || (v2 == redv[t] && i2 < redi[t])) {
                    redv[t] = v2; redi[t] = i2;
                }
            }
            __syncthreads();
        }
        if (t == 0) { IDX[blockIdx.x * KNN + k] = redi[0]; vals[redi[0]] = -FLT_BIG; }
        __syncthreads();
    }
}

// ---------------------------------------------------------------------------
// Fused P/Q GEMM via WMMA f32:
//   P = X (4096 x cin) @ Wa^T   , Wa   = W[:, :cin]
//   Q = X @ (Wb - Wa)^T         , Wb   = W[:, cin:]
// Shares the A operand across two v_wmma per K-step. blockDim = (32,4).
// ---------------------------------------------------------------------------
__global__ void k_pq_gemm(const float* __restrict__ X, const float* __restrict__ W,
                          float* __restrict__ P, float* __restrict__ Q,
                          int cin, int cout) {
    const int lane = threadIdx.x;
    const int half = lane >> 4;
    const int lm   = lane & 15;
    const int m0 = blockIdx.x * 16;
    const int n0 = (blockIdx.y * 4 + threadIdx.y) * 16;

    const float* xrow = X + (size_t)(m0 + lm) * cin;
    const float* wrow = W + (size_t)(n0 + lm) * (2 * cin);

    v8f accP = {};
    v8f accQ = {};
    for (int kk = 0; kk < cin; kk += 4) {
        int k0 = kk + 2 * half;
        v2f a;  a.x  = xrow[k0];           a.y  = xrow[k0 + 1];
        v2f bp; bp.x = wrow[k0];           bp.y = wrow[k0 + 1];
        v2f bq; bq.x = wrow[cin + k0] - bp.x;
                bq.y = wrow[cin + k0 + 1] - bp.y;
        accP = __builtin_amdgcn_wmma_f32_16x16x4_f32(false, a, false, bp,
                                                     (short)0, accP, false, false);
        accQ = __builtin_amdgcn_wmma_f32_16x16x4_f32(false, a, false, bq,
                                                     (short)0, accQ, false, false);
    }
#pragma unroll
    for (int r = 0; r < 8; ++r) {
        int row = m0 + r + 8 * half;
        P[(size_t)row * cout + n0 + lm] = accP[r];
        Q[(size_t)row * cout + n0 + lm] = accQ[r];
    }
}

// ---------------------------------------------------------------------------
// Pass 1: per-channel BN statistics of h[b,n,k,o] = P[nbr][o] + Q[bn][o].
// blockDim = (64,4); each thread owns one channel, strides over rows.
// ---------------------------------------------------------------------------
__global__ void k_gather_stats(const float* __restrict__ P, const float* __restrict__ Q,
                               const int* __restrict__ IDX,
                               float* __restrict__ ssum, float* __restrict__ ssq,
                               int cout) {
    const int o = blockIdx.x * 64 + threadIdx.x;
    float s = 0.f, q2 = 0.f;
    for (int r = blockIdx.y * 4 + threadIdx.y; r < NROWS; r += gridDim.y * 4) {
        const int base = r & ~(NPTS - 1);
        const float qv = Q[(size_t)r * cout + o];
        const int* ix = IDX + r * KNN;
        for (int k = 0; k < KNN; ++k) {
            float v = P[(size_t)(base + ix[k]) * cout + o] + qv;
            s += v; q2 += v * v;
        }
    }
    atomicAdd(&ssum[o], s);
    atomicAdd(&ssq[o], q2);
}

// ---------------------------------------------------------------------------
// Pass 2: BN affine + leaky + max over K (min if scale<0), write next-layer X.
// One block per (b,n) row; 256 threads stride channels.
// ---------------------------------------------------------------------------
__global__ void k_gather_finalize(const float* __restrict__ P, const float* __restrict__ Q,
                                  const int* __restrict__ IDX,
                                  const float* __restrict__ ssum, const float* __restrict__ ssq,
                                  const float* __restrict__ gamma, const float* __restrict__ beta,
                                  float* __restrict__ Xn, int cout) {
    __shared__ int ix[KNN];
    const int r = blockIdx.x;
    const int base = r & ~(NPTS - 1);
    if (threadIdx.x < KNN) ix[threadIdx.x] = IDX[r * KNN + threadIdx.x];
    __syncthreads();
    const float inv = 1.0f / (float)BNK;
    for (int o = threadIdx.x; o < cout; o += 256) {
        float mean = ssum[o] * inv;
        float var  = ssq[o] * inv - mean * mean;
        float sc = gamma[o] * rsqrtf(var + 1e-5f);
        float sh = beta[o] - mean * sc;
        float qv = Q[(size_t)r * cout + o];
        float m;
        if (sc >= 0.f) {
            m = -FLT_BIG;
            for (int k = 0; k < KNN; ++k)
                m = fmaxf(m, P[(size_t)(base + ix[k]) * cout + o]);
        } else {
            m = FLT_BIG;
            for (int k = 0; k < KNN; ++k)
                m = fminf(m, P[(size_t)(base + ix[k]) * cout + o]);
        }
        float h = sc * (m + qv) + sh;
        Xn[(size_t)r * cout + o] = (h > 0.f) ? h : 0.2f * h;
    }
}

// ---------------------------------------------------------------------------
// Per-layer global pooling: p1 = max over n, p2 = mean over n.
// grid = (cout/64, B), blockDim = 64.
// ---------------------------------------------------------------------------
__global__ void k_pool(const float* __restrict__ Xn, float* __restrict__ p1,
                       float* __restrict__ p2, int cout, int off) {
    const int o = blockIdx.x * 64 + threadIdx.x;
    const int b = blockIdx.y;
    const float* Xb = Xn + (size_t)b * NPTS * cout;
    float mx = -FLT_BIG, s = 0.f;
    for (int n = 0; n < NPTS; ++n) {
        float v = Xb[(size_t)n * cout + o];
        mx = fmaxf(mx, v); s += v;
    }
    p1[b * 2048 + off + o] = mx;
    p2[b * 2048 + off + o] = s * (1.0f / (float)NPTS);
}

// ---------------------------------------------------------------------------
// z[b][e] = [p1[b], p2[b]] . Wm[e] + bm[e].  One block per e, 128 threads.
// ---------------------------------------------------------------------------
__global__ void k_mlp(const float* __restrict__ p1, const float* __restrict__ p2,
                      const float* __restrict__ Wm, const float* __restrict__ bm,
                      float* __restrict__ z) {
    __shared__ float red[4][128];
    const int e = blockIdx.x;
    const int t = threadIdx.x;
    const float* w = Wm + (size_t)e * 4096;
    float acc[4] = {0.f, 0.f, 0.f, 0.f};
    for (int c = t; c < 2048; c += 128) {
        float w1 = w[c], w2 = w[2048 + c];
#pragma unroll
        for (int b = 0; b < 4; ++b)
            acc[b] += p1[b * 2048 + c] * w1 + p2[b * 2048 + c] * w2;
    }
#pragma unroll
    for (int b = 0; b < 4; ++b) red[b][t] = acc[b];
    __syncthreads();
    for (int s = 64; s > 0; s >>= 1) {
        if (t < s)
#pragma unroll
            for (int b = 0; b < 4; ++b) red[b][t] += red[b][t + s];
        __syncthreads();
    }
    if (t == 0) {
#pragma unroll
        for (int b = 0; b < 4; ++b) z[b * 512 + e] = red[b][0] + bm[e];
    }
}

// ---------------------------------------------------------------------------
// Final batch-norm over batch dim (4 samples) + leaky relu.
// ---------------------------------------------------------------------------
__global__ void k_final_bn(const float* __restrict__ z, const float* __restrict__ gm,
                           const float* __restrict__ betam, float* __restrict__ out) {
    int e = blockIdx.x * 256 + threadIdx.x;
    if (e >= 512) return;
    float z0 = z[e], z1 = z[512 + e], z2 = z[1024 + e], z3 = z[1536 + e];
    float mean = 0.25f * (z0 + z1 + z2 + z3);
    float var  = 0.25f * (z0 * z0 + z1 * z1 + z2 * z2 + z3 * z3) - mean * mean;
    float sc = gm[e] * rsqrtf(var + 1e-5f);
    float sh = betam[e] - mean * sc;
#pragma unroll
    for (int b = 0; b < 4; ++b) {
        float v = sc * z[b * 512 + e] + sh;
        out[b * 512 + e] = (v > 0.f) ? v : 0.2f * v;
    }
}

__global__ void k_zero(float* __restrict__ p, int n) {
    int i = blockIdx.x * 256 + threadIdx.x;
    if (i < n) p[i] = 0.f;
}

// ---------------------------------------------------------------------------
extern "C" void kernel_launch(void* const* d_in, const int* in_sizes, int n_in,
                              void* d_out, int out_size, void* d_ws, size_t ws_size,
                              hipStream_t stream) {
    (void)in_sizes; (void)n_in; (void)out_size; (void)ws_size;

    const float* x = (const float*)d_in[0];
    const float* Wm    = (const float*)d_in[19];
    const float* bm    = (const float*)d_in[20];
    const float* gm    = (const float*)d_in[21];
    const float* betam = (const float*)d_in[22];
    float* out = (float*)d_out;

    // Workspace layout (floats); ~84.3 MB total, fully L2-resident on MI455X.
    const size_t BIG = (size_t)NROWS * 1024;       // 4 Mi floats
    float* ws   = (float*)d_ws;
    float* Xa   = ws;
    float* Xb   = Xa + BIG;
    float* P    = Xb + BIG;
    float* Q    = P  + BIG;
    float* PD   = Q  + BIG;
    float* xx   = PD + BIG;          // 4096
    float* ssum = xx + NROWS;        // 1024
    float* ssq  = ssum + 1024;       // 1024
    float* p1   = ssq + 1024;        // 4*2048
    float* p2   = p1 + 4 * 2048;     // 4*2048
    float* z    = p2 + 4 * 2048;     // 4*512
    int*   IDX  = (int*)(z + 4 * 512);  // 4096*20 ints

    static const int cins[6]  = {8, 64, 64, 128, 256, 512};
    static const int couts[6] = {64, 64, 128, 256, 512, 1024};

    // x (4,8,1024) -> (B*N, 8)
    k_transpose_in<<<(BATCH * 8 * NPTS + 255) / 256, 256, 0, stream>>>(x, Xa);

    float* Xcur = Xa;
    float* Xnxt = Xb;
    int off = 0;
    for (int l = 0; l < 6; ++l) {
        const int C = cins[l];
        const int O = couts[l];
        const float* W = (const float*)d_in[1 + 3 * l];
        const float* g = (const float*)d_in[2 + 3 * l];
        const float* bt = (const float*)d_in[3 + 3 * l];

        k_row_norms<<<(NROWS + 255) / 256, 256, 0, stream>>>(Xcur, xx, C);

        k_gram_pd<<<dim3(NPTS / 16, NPTS / 64, BATCH), dim3(32, 4), 0, stream>>>(
            Xcur, xx, PD, C);

        k_topk<<<NROWS, 256, 0, stream>>>(PD, IDX);

        k_pq_gemm<<<dim3(NROWS / 16, O / 64), dim3(32, 4), 0, stream>>>(
            Xcur, W, P, Q, C, O);

        k_zero<<<(2 * O + 255) / 256, 256, 0, stream>>>(ssum, 2 * O);  // ssum+ssq contiguous

        k_gather_stats<<<dim3(O / 64, 32), dim3(64, 4), 0, stream>>>(
            P, Q, IDX, ssum, ssq, O);

        k_gather_finalize<<<NROWS, 256, 0, stream>>>(
            P, Q, IDX, ssum, ssq, g, bt, Xnxt, O);

        k_pool<<<dim3(O / 64, BATCH), 64, 0, stream>>>(Xnxt, p1, p2, O, off);
        off += O;

        float* tmp = Xcur; Xcur = Xnxt; Xnxt = tmp;
    }

    k_mlp<<<512, 128, 0, stream>>>(p1, p2, Wm, bm, z);
    k_final_bn<<<2, 256, 0, stream>>>(z, gm, betam, out);
}